// DGLMPNNLayer_88347477279350
// MI455X (gfx1250) — compile-verified
//
#include <hip/hip_runtime.h>

#define N_NODES 10000
#define N_EDGES 50000
#define HID 64
#define EDGE_DIM 16

typedef float v2f __attribute__((ext_vector_type(2)));
typedef float v8f __attribute__((ext_vector_type(8)));

#define KPAIRS 32          // 64 K-rows (h) stored as 32 interleaved pairs
#define NCOLS 1088         // 16*64 (d*64+o) + 64 bias columns
#define LDS_STRIDE 1104    // padded: (1104*2)%64 == 32 -> two half-waves hit disjoint LDS banks
#define NTILES (N_EDGES / 16)

// out[n,h] = nf[n,h] + bias[h]  (residual + bias folded into the scatter target)
__global__ void mpnn_init_out(const float* __restrict__ nf,
                              const float* __restrict__ bias,
                              float* __restrict__ out) {
  int i = blockIdx.x * blockDim.x + threadIdx.x;
  if (i < N_NODES * HID) out[i] = nf[i] + bias[i & (HID - 1)];
}

__global__ void __launch_bounds__(128)
mpnn_edge_kernel(const float* __restrict__ nf,
                 const float* __restrict__ ef,
                 const int* __restrict__ src,
                 const int* __restrict__ dst,
                 const float* __restrict__ W_edge,
                 const float* __restrict__ b_edge,
                 float* __restrict__ out) {
  extern __shared__ char smem_raw[];
  v2f* __restrict__ ldsw = (v2f*)smem_raw;

  const int tid = threadIdx.x;

  // ---- Stage M_aug (K=h in [0,64), N=c in [0,1088)) into LDS, K-pair interleaved:
  //      ldsw[kp*STRIDE + c] = { M[2kp, c], M[2kp+1, c] }
  //      M[h, c] = c<1024 ? W_edge[(c>>6)*4096 + h*64 + (c&63)] : b_edge[h*64 + (c-1024)]
  for (int idx = tid; idx < KPAIRS * NCOLS; idx += blockDim.x) {
    int kp = idx / NCOLS;
    int c  = idx - kp * NCOLS;
    int h0 = kp << 1;
    float m0, m1;
    if (c < 1024) {
      int d = c >> 6, o = c & 63;
      m0 = W_edge[d * 4096 + h0 * 64 + o];
      m1 = W_edge[d * 4096 + (h0 + 1) * 64 + o];
    } else {
      int o = c - 1024;
      m0 = b_edge[h0 * 64 + o];
      m1 = b_edge[(h0 + 1) * 64 + o];
    }
    v2f v; v.x = m0; v.y = m1;
    ldsw[kp * LDS_STRIDE + c] = v;
  }
  __syncthreads();

  const int lane = tid & 31;   // wave32
  const int wid  = tid >> 5;
  const int hi   = lane >> 4;  // half-wave: selects K offset (+2) in f32 WMMA fragments
  const int lo   = lane & 15;
  const int nwaves = (blockDim.x >> 5) * gridDim.x;
  const int wslot  = blockIdx.x * (blockDim.x >> 5) + wid;
  const v2f* __restrict__ nf2 = (const v2f*)nf;

  const v8f vzero = {0.f, 0.f, 0.f, 0.f, 0.f, 0.f, 0.f, 0.f};

  for (int tile = wslot; tile < NTILES; tile += nwaves) {
    const int e0 = tile << 4;

    // ---- A fragments: X_tile(16x64) = nf[src[e0..e0+15]] in 16x4-f32 A layout.
    //      lane<16 holds (M=lo, K=4kk,4kk+1); lane>=16 holds (M=lo, K=4kk+2,4kk+3)
    const int row = src[e0 + lo];
    v2f a[16];
#pragma unroll
    for (int kk = 0; kk < 16; ++kk)
      a[kk] = nf2[row * 32 + (kk << 1) + hi];

    // ---- ef stash for ds_bpermute broadcast:
    //      efr[r] on lane L = ef[e0 + r + 8*hi(L)][L&15]
    float efr[8];
#pragma unroll
    for (int r = 0; r < 8; ++r)
      efr[r] = ef[(e0 + r + (hi << 3)) * EDGE_DIM + lo];

    // msg accumulators: 4 o-blocks of 16 cols; C/D layout (lane->N, vgpr r -> M=r+8*hi)
    v8f macc[4];
#pragma unroll
    for (int jb = 0; jb < 4; ++jb) macc[jb] = vzero;

    // ---- 68 N-tiles of Z_aug = X_tile @ M_aug; K=64 -> 16 wmma_f32_16x16x4 each
    for (int j = 0; j < 68; ++j) {
      const int c = (j << 4) + lo;
      const v2f* wp = ldsw + (hi * LDS_STRIDE + c);
      v8f z = vzero;
#pragma unroll
      for (int kk = 0; kk < 16; ++kk) {
        v2f b = wp[(kk << 1) * LDS_STRIDE];  // K rows {4kk+2hi, 4kk+2hi+1}, col c
#if __has_builtin(__builtin_amdgcn_wmma_f32_16x16x4_f32)
        z = __builtin_amdgcn_wmma_f32_16x16x4_f32(false, a[kk], false, b,
                                                  (short)0, z, false, false);
#else
        // fallback so the TU still compiles if the builtin name differs
        z[0] += a[kk].x * b.x + a[kk].y * b.y;
#endif
      }
      if (j < 64) {
        // msg[:, o-block j&3] += ef[:, j>>2] * Ztile   (scale depends on edge row M)
        const int d    = j >> 2;
        const int jb   = j & 3;
        const int bidx = (d | (lane & 16)) << 2;  // src lane = d + 16*hi, byte index
#pragma unroll
        for (int r = 0; r < 8; ++r) {
          int si = __builtin_amdgcn_ds_bpermute(bidx, __builtin_bit_cast(int, efr[r]));
          macc[jb][r] += __builtin_bit_cast(float, si) * z[r];
        }
      } else {
        // bias columns: msg += X_tile @ B (unscaled)
        const int jb = j - 64;
#pragma unroll
        for (int r = 0; r < 8; ++r) macc[jb][r] += z[r];
      }
    }

    // ---- scatter: lane holds (M = r + 8*hi, N = lo) of msg[16x64]
#pragma unroll
    for (int r = 0; r < 8; ++r) {
      const int drow = dst[e0 + r + (hi << 3)];
      float* op = out + drow * HID + lo;
#pragma unroll
      for (int jb = 0; jb < 4; ++jb)
        atomicAdd(op + (jb << 4), macc[jb][r]);
    }
  }
}

extern "C" void kernel_launch(void* const* d_in, const int* in_sizes, int n_in,
                              void* d_out, int out_size, void* d_ws, size_t ws_size,
                              hipStream_t stream) {
  const float* nf     = (const float*)d_in[0];
  const float* ef     = (const float*)d_in[1];
  const int*   src    = (const int*)d_in[2];
  const int*   dst    = (const int*)d_in[3];
  const float* W_edge = (const float*)d_in[4];
  const float* b_edge = (const float*)d_in[5];
  const float* bias   = (const float*)d_in[6];
  float* out = (float*)d_out;

  mpnn_init_out<<<(N_NODES * HID + 255) / 256, 256, 0, stream>>>(nf, bias, out);

  const size_t lds_bytes = (size_t)KPAIRS * LDS_STRIDE * sizeof(v2f);  // 282,624 B < 320 KB
  mpnn_edge_kernel<<<160, 128, lds_bytes, stream>>>(nf, ef, src, dst, W_edge, b_edge, out);
}